// ResidualGATBlock_10110353015046
// MI455X (gfx1250) — compile-verified
//
#include <hip/hip_runtime.h>
#include <math.h>

#define N_NODES 40000
#define E_EDGES 640000
#define ET      (E_EDGES + N_NODES)   // 680000 incl. self loops
#define DIM     128
#define HEADS   8
#define HD      16
#define NEG_SLOPE 0.2f

typedef float v2f __attribute__((ext_vector_type(2)));
typedef float v8f __attribute__((ext_vector_type(8)));

// ---------------------------------------------------------------------------
// Kernel 1: fused xs = x@W_src + b_src ; xd = x@W_dst + b_dst
// fp32 WMMA (V_WMMA_F32_16X16X4_F32), one 16-row M-tile per block,
// 8 waves -> 8 N-tiles, each wave accumulates both weight matrices
// sharing the A fragments staged in LDS.
// ---------------------------------------------------------------------------
__global__ __launch_bounds__(256) void gemm_xsxd(
    const float* __restrict__ x,
    const float* __restrict__ Ws, const float* __restrict__ bs,
    const float* __restrict__ Wd, const float* __restrict__ bd,
    float* __restrict__ xs, float* __restrict__ xd)
{
    __shared__ float tile[16 * DIM];            // 8 KB x-tile
    const int tid   = threadIdx.x;
    const int mbase = blockIdx.x * 16;

    // stage 16x128 tile of x into LDS (float4 vectorized)
    {
        const float4* gx = reinterpret_cast<const float4*>(x + (size_t)mbase * DIM);
        float4* lx = reinterpret_cast<float4*>(tile);
        #pragma unroll
        for (int i = tid; i < 16 * DIM / 4; i += 256) lx[i] = gx[i];
    }
    __syncthreads();

    const int wave = tid >> 5;
    const int lane = tid & 31;
    const int half = lane >> 4;          // which K-half of the fragment
    const int r    = lane & 15;          // A: row M ; B: column N
    const int ncol = wave * 16 + r;      // global output column

    v8f cs = {0.f,0.f,0.f,0.f,0.f,0.f,0.f,0.f};
    v8f cd = {0.f,0.f,0.f,0.f,0.f,0.f,0.f,0.f};

    for (int kk = 0; kk < DIM / 4; ++kk) {
        const int k0 = kk * 4 + half * 2;          // K = 2*half + component
        v2f a;
        a.x = tile[r * DIM + k0];
        a.y = tile[r * DIM + k0 + 1];
        v2f b1;
        b1.x = Ws[(size_t)k0 * DIM + ncol];
        b1.y = Ws[(size_t)(k0 + 1) * DIM + ncol];
        v2f b2;
        b2.x = Wd[(size_t)k0 * DIM + ncol];
        b2.y = Wd[(size_t)(k0 + 1) * DIM + ncol];
        cs = __builtin_amdgcn_wmma_f32_16x16x4_f32(false, a, false, b1, (short)0, cs, false, false);
        cd = __builtin_amdgcn_wmma_f32_16x16x4_f32(false, a, false, b2, (short)0, cd, false, false);
    }

    const float biass = bs[ncol];
    const float biasd = bd[ncol];
    #pragma unroll
    for (int j = 0; j < 8; ++j) {                  // C/D: VGPR j -> row j + 8*half
        const int row = mbase + j + 8 * half;
        xs[(size_t)row * DIM + ncol] = cs[j] + biass;
        xd[(size_t)row * DIM + ncol] = cd[j] + biasd;
    }
}

// ---------------------------------------------------------------------------
// float atomic max via int ordering trick (init value must be -inf)
// ---------------------------------------------------------------------------
__device__ inline void atomicMaxFloat(float* addr, float val)
{
    if (val >= 0.0f) atomicMax(reinterpret_cast<int*>(addr), __float_as_int(val));
    else             atomicMin(reinterpret_cast<unsigned int*>(addr), __float_as_uint(val));
}

// ---------------------------------------------------------------------------
// Kernel 0: init maxlog = -inf, denom = 0, out accumulator = 0
// ---------------------------------------------------------------------------
__global__ __launch_bounds__(256) void init_buffers(
    float* __restrict__ maxlog, float* __restrict__ denom, float* __restrict__ out)
{
    const long long i = (long long)blockIdx.x * 256 + threadIdx.x;   // grid covers N*DIM
    out[i] = 0.0f;
    if (i < (long long)N_NODES * HEADS) {
        maxlog[i] = -INFINITY;
        denom[i]  = 0.0f;
    }
}

// ---------------------------------------------------------------------------
// Kernel 2: per-edge GATv2 logits + segment max. One wave per edge;
// each lane handles 4 consecutive channels (float4), heads = groups of 4 lanes.
// ---------------------------------------------------------------------------
__global__ __launch_bounds__(256) void edge_logits(
    const float* __restrict__ xs, const float* __restrict__ xd,
    const int* __restrict__ ei,  const float* __restrict__ att,
    float* __restrict__ logcache, float* __restrict__ maxlog)
{
    const int wave = threadIdx.x >> 5;
    const int lane = threadIdx.x & 31;
    const long long e = (long long)blockIdx.x * 8 + wave;   // grid exact: < ET

    int s, d;
    if (e < E_EDGES) { s = ei[e]; d = ei[E_EDGES + e]; }
    else             { s = (int)(e - E_EDGES); d = s; }     // self loop

    const int c0 = lane * 4;
    const float4 a  = *reinterpret_cast<const float4*>(xs + (size_t)s * DIM + c0);
    const float4 b  = *reinterpret_cast<const float4*>(xd + (size_t)d * DIM + c0);
    const float4 at = *reinterpret_cast<const float4*>(att + c0);

    float p = 0.0f, v;
    v = a.x + b.x; v = (v >= 0.0f) ? v : NEG_SLOPE * v; p += v * at.x;
    v = a.y + b.y; v = (v >= 0.0f) ? v : NEG_SLOPE * v; p += v * at.y;
    v = a.z + b.z; v = (v >= 0.0f) ? v : NEG_SLOPE * v; p += v * at.z;
    v = a.w + b.w; v = (v >= 0.0f) ? v : NEG_SLOPE * v; p += v * at.w;

    // reduce over the 4 lanes of each head (lanes 4h..4h+3)
    p += __shfl_xor(p, 1);
    p += __shfl_xor(p, 2);

    if ((lane & 3) == 0) {
        const int h = lane >> 2;
        logcache[e * HEADS + h] = p;
        atomicMaxFloat(&maxlog[(size_t)d * HEADS + h], p);
    }
}

// ---------------------------------------------------------------------------
// Kernel 3: ex = exp(logit - max[dst]); denom[dst] += ex. Thread per (edge,head).
// ---------------------------------------------------------------------------
__global__ __launch_bounds__(256) void edge_softmax_denom(
    const int* __restrict__ ei, float* __restrict__ logcache,
    const float* __restrict__ maxlog, float* __restrict__ denom)
{
    const long long idx = (long long)blockIdx.x * 256 + threadIdx.x;  // grid exact: < ET*8
    const long long e   = idx >> 3;
    const int       h   = (int)(idx & 7);

    int d;
    if (e < E_EDGES) d = ei[E_EDGES + e];
    else             d = (int)(e - E_EDGES);

    const float ex = expf(logcache[idx] - maxlog[(size_t)d * HEADS + h]);
    logcache[idx] = ex;                         // cache numerator for pass 3
    atomicAdd(&denom[(size_t)d * HEADS + h], ex);
}

// ---------------------------------------------------------------------------
// Kernel 4: weighted scatter-add of source features. One wave per edge.
// ---------------------------------------------------------------------------
__global__ __launch_bounds__(256) void edge_aggregate(
    const float* __restrict__ xs, const int* __restrict__ ei,
    const float* __restrict__ excache, const float* __restrict__ denom,
    float* __restrict__ out)
{
    const int wave = threadIdx.x >> 5;
    const int lane = threadIdx.x & 31;
    const long long e = (long long)blockIdx.x * 8 + wave;

    int s, d;
    if (e < E_EDGES) { s = ei[e]; d = ei[E_EDGES + e]; }
    else             { s = (int)(e - E_EDGES); d = s; }

    const int h  = lane >> 2;
    const int c0 = lane * 4;
    const float alpha = excache[e * HEADS + h] / denom[(size_t)d * HEADS + h];

    const float4 a = *reinterpret_cast<const float4*>(xs + (size_t)s * DIM + c0);
    float* o = out + (size_t)d * DIM + c0;
    atomicAdd(o + 0, a.x * alpha);
    atomicAdd(o + 1, a.y * alpha);
    atomicAdd(o + 2, a.z * alpha);
    atomicAdd(o + 3, a.w * alpha);
}

// ---------------------------------------------------------------------------
// Kernel 5: residual + LayerNorm + exact GELU. One wave per node (32 lanes x 4 ch).
// ---------------------------------------------------------------------------
__global__ __launch_bounds__(256) void finalize(
    const float* __restrict__ x, const float* __restrict__ bias_out,
    const float* __restrict__ gamma, const float* __restrict__ beta,
    float* __restrict__ out)
{
    const int wave = threadIdx.x >> 5;
    const int lane = threadIdx.x & 31;
    const int n    = blockIdx.x * 8 + wave;      // grid exact: < N
    const int c0   = lane * 4;

    const float4 acc = *reinterpret_cast<const float4*>(out      + (size_t)n * DIM + c0);
    const float4 xv  = *reinterpret_cast<const float4*>(x        + (size_t)n * DIM + c0);
    const float4 bo  = *reinterpret_cast<const float4*>(bias_out + c0);

    float h0 = acc.x + bo.x + xv.x;
    float h1 = acc.y + bo.y + xv.y;
    float h2 = acc.z + bo.z + xv.z;
    float h3 = acc.w + bo.w + xv.w;

    float s = h0 + h1 + h2 + h3;
    float q = h0*h0 + h1*h1 + h2*h2 + h3*h3;
    #pragma unroll
    for (int off = 16; off >= 1; off >>= 1) {
        s += __shfl_xor(s, off);
        q += __shfl_xor(q, off);
    }
    const float mean = s * (1.0f / DIM);
    const float var  = q * (1.0f / DIM) - mean * mean;
    const float rstd = rsqrtf(var + 1e-5f);

    const float4 g = *reinterpret_cast<const float4*>(gamma + c0);
    const float4 bb = *reinterpret_cast<const float4*>(beta + c0);

    float4 r;
    float t;
    t = (h0 - mean) * rstd * g.x + bb.x; r.x = 0.5f * t * (1.0f + erff(t * 0.70710678118f));
    t = (h1 - mean) * rstd * g.y + bb.y; r.y = 0.5f * t * (1.0f + erff(t * 0.70710678118f));
    t = (h2 - mean) * rstd * g.z + bb.z; r.z = 0.5f * t * (1.0f + erff(t * 0.70710678118f));
    t = (h3 - mean) * rstd * g.w + bb.w; r.w = 0.5f * t * (1.0f + erff(t * 0.70710678118f));

    *reinterpret_cast<float4*>(out + (size_t)n * DIM + c0) = r;
}

// ---------------------------------------------------------------------------
extern "C" void kernel_launch(void* const* d_in, const int* in_sizes, int n_in,
                              void* d_out, int out_size, void* d_ws, size_t ws_size,
                              hipStream_t stream)
{
    const float* x     = (const float*)d_in[0];
    const int*   ei    = (const int*)  d_in[1];   // edge_index, [2, E] flat
    const float* Ws    = (const float*)d_in[2];
    const float* bs    = (const float*)d_in[3];
    const float* Wd    = (const float*)d_in[4];
    const float* bd    = (const float*)d_in[5];
    const float* att   = (const float*)d_in[6];
    const float* bo    = (const float*)d_in[7];
    const float* gamma = (const float*)d_in[8];
    const float* beta  = (const float*)d_in[9];
    float* out = (float*)d_out;

    // workspace layout (floats): xs | xd | maxlog | denom | excache  (~65.3 MB)
    float* xs      = (float*)d_ws;
    float* xd      = xs + (size_t)N_NODES * DIM;
    float* maxlog  = xd + (size_t)N_NODES * DIM;
    float* denom   = maxlog + (size_t)N_NODES * HEADS;
    float* excache = denom  + (size_t)N_NODES * HEADS;

    // grids are exact: N*DIM/256 = 20000, N/16 = 2500, ET/8 = 85000,
    // ET*HEADS/256 = 21250, N/8 = 5000
    init_buffers      <<<20000, 256, 0, stream>>>(maxlog, denom, out);
    gemm_xsxd         <<< 2500, 256, 0, stream>>>(x, Ws, bs, Wd, bd, xs, xd);
    edge_logits       <<<85000, 256, 0, stream>>>(xs, xd, ei, att, excache, maxlog);
    edge_softmax_denom<<<21250, 256, 0, stream>>>(ei, excache, maxlog, denom);
    edge_aggregate    <<<85000, 256, 0, stream>>>(xs, ei, excache, denom, out);
    finalize          <<< 5000, 256, 0, stream>>>(x, bo, gamma, beta, out);
}